// VisionMambaEncoder2D_14121852470055
// MI455X (gfx1250) — compile-verified
//
#include <hip/hip_runtime.h>
#include <cmath>

// ---------------------------------------------------------------------------
// VisionMamba2D encoder for gfx1250 (MI455X), fp32 end-to-end.
// GEMMs use V_WMMA_F32_16X16X4_F32 (exact fp32 matrix path, wave32),
// 32x64 tile per wave with clause-grouped b64 loads overlapping WMMAs.
// Long (L=4096) selective scans use a 3-phase chunked parallel scan
// (64-step chunks) to break the serial critical path; short (L=64) scans
// run directly. State (16-wide) lives in registers throughout.
//
// Input flattening assumption (setup_inputs dict insertion order, leaves in
// nested-dict insertion order):
//   0: x
//   1..16: ln_g, ln_b, tox_w, tox_b, toz_w, toz_b, cfh_w, cfh_b, cbh_w, cbh_b,
//          cfv_w, cfv_b, cbv_w, cbv_b, out_w, out_b
//   17+9*dir: in_proj_w, conv1d_w, conv1d_b, x_proj_w, dt_proj_w, dt_proj_b,
//             A_log, D, out_proj_w      (dir: 0=fh, 1=bh, 2=fv, 3=bv)
// ---------------------------------------------------------------------------

#define BATCH 2
#define CH    128
#define HH    64
#define WWID  64
#define HWSZ  (HH * WWID)        // 4096
#define NTOK  (BATCH * HWSZ)     // 8192
#define DI    256
#define DS    16
#define DTR   8
#define XD    (DTR + 2 * DS)     // 40
#define XZLD  (2 * DI)           // 512
#define CHUNK 64                 // parallel-scan chunk length

typedef float v2f __attribute__((ext_vector_type(2)));
typedef float v8f __attribute__((ext_vector_type(8)));

enum { EPI_NONE = 0, EPI_BIAS = 1, EPI_GELU = 2, EPI_SOFTPLUS = 3 };

__device__ __forceinline__ float apply_epi(int EPI, float v) {
    if (EPI == EPI_GELU)     v = 0.5f * v * (1.f + erff(v * 0.7071067811865475f));
    if (EPI == EPI_SOFTPLUS) v = (v > 20.f) ? v : log1pf(expf(v));
    return v;
}

#define WMMA_F32(a, b, c) \
    __builtin_amdgcn_wmma_f32_16x16x4_f32(false, (a), false, (b), (short)0, (c), false, false)

// ---------------------------------------------------------------------------
// Fast WMMA GEMM: Co[M,N] = epi(A[M,K] @ Wt[N,K]^T + bias[N])
// One wave per 32(M) x 64(N) tile. Requires: M % 32 == 0, N % 64 == 0,
// K % 4 == 0 and K >= 4. Fragments for k+4 are loaded before the 8 WMMAs
// for k so the clause of b64 loads overlaps the matrix pipe.
// ---------------------------------------------------------------------------
template <int EPI>
__global__ __launch_bounds__(32)
void gemm_wmma2_kernel(const float* __restrict__ A, int lda,
                       const float* __restrict__ Wt, int ldw,
                       const float* __restrict__ bias,
                       float* __restrict__ Co, int ldc, int K)
{
    const int lane = threadIdx.x;
    const int half = lane >> 4;      // 0 or 1
    const int lm   = lane & 15;
    const int m0   = blockIdx.y << 5;    // 32 rows
    const int n0   = blockIdx.x << 6;    // 64 cols

    // A fragment rows: lane<16 -> K pair {k,k+1}, lane>=16 -> {k+2,k+3}
    const float* __restrict__ Ap0 = A + (size_t)(m0 + lm) * lda + (half << 1);
    const float* __restrict__ Ap1 = Ap0 + (size_t)16 * lda;
    const float* __restrict__ W0  = Wt + (size_t)(n0 + lm) * ldw + (half << 1);
    const float* __restrict__ W1  = W0 + (size_t)16 * ldw;
    const float* __restrict__ W2  = W0 + (size_t)32 * ldw;
    const float* __restrict__ W3  = W0 + (size_t)48 * ldw;

    v8f c00 = {}, c01 = {}, c02 = {}, c03 = {};
    v8f c10 = {}, c11 = {}, c12 = {}, c13 = {};

    // pipeline prologue: fragments for k = 0
    v2f a0 = *(const v2f*)(Ap0);
    v2f a1 = *(const v2f*)(Ap1);
    v2f b0 = *(const v2f*)(W0);
    v2f b1 = *(const v2f*)(W1);
    v2f b2 = *(const v2f*)(W2);
    v2f b3 = *(const v2f*)(W3);

    for (int k0 = 4; k0 < K; k0 += 4) {
        // issue next-step loads first: they overlap the WMMAs below
        const v2f a0n = *(const v2f*)(Ap0 + k0);
        const v2f a1n = *(const v2f*)(Ap1 + k0);
        const v2f b0n = *(const v2f*)(W0 + k0);
        const v2f b1n = *(const v2f*)(W1 + k0);
        const v2f b2n = *(const v2f*)(W2 + k0);
        const v2f b3n = *(const v2f*)(W3 + k0);

        c00 = WMMA_F32(a0, b0, c00);
        c01 = WMMA_F32(a0, b1, c01);
        c02 = WMMA_F32(a0, b2, c02);
        c03 = WMMA_F32(a0, b3, c03);
        c10 = WMMA_F32(a1, b0, c10);
        c11 = WMMA_F32(a1, b1, c11);
        c12 = WMMA_F32(a1, b2, c12);
        c13 = WMMA_F32(a1, b3, c13);

        a0 = a0n; a1 = a1n;
        b0 = b0n; b1 = b1n; b2 = b2n; b3 = b3n;
    }
    // epilogue k-step
    c00 = WMMA_F32(a0, b0, c00);
    c01 = WMMA_F32(a0, b1, c01);
    c02 = WMMA_F32(a0, b2, c02);
    c03 = WMMA_F32(a0, b3, c03);
    c10 = WMMA_F32(a1, b0, c10);
    c11 = WMMA_F32(a1, b1, c11);
    c12 = WMMA_F32(a1, b2, c12);
    c13 = WMMA_F32(a1, b3, c13);

    v8f accs[2][4] = {{c00, c01, c02, c03}, {c10, c11, c12, c13}};
    #pragma unroll
    for (int mt = 0; mt < 2; ++mt) {
        #pragma unroll
        for (int t = 0; t < 4; ++t) {
            const int n = n0 + (t << 4) + lm;
            float bv = 0.f;
            if (EPI != EPI_NONE) bv = bias[n];
            #pragma unroll
            for (int r = 0; r < 8; ++r) {
                // C/D layout: M = r + 8*half within each 16-row tile
                const int m = m0 + (mt << 4) + r + (half << 3);
                Co[(size_t)m * ldc + n] = apply_epi(EPI, accs[mt][t][r] + bv);
            }
        }
    }
}

// ---------------------------------------------------------------------------
// Guarded WMMA GEMM for N not a multiple of 64 (x_proj, N=40).
// One wave per 16x64 tile; B loads and C stores bounds-checked against N.
// ---------------------------------------------------------------------------
template <int EPI>
__global__ __launch_bounds__(32)
void gemm_wmma_guard_kernel(const float* __restrict__ A, int lda,
                            const float* __restrict__ Wt, int ldw,
                            const float* __restrict__ bias,
                            float* __restrict__ Co, int ldc,
                            int N, int K)
{
    const int lane = threadIdx.x;
    const int half = lane >> 4;
    const int lm   = lane & 15;
    const int m0   = blockIdx.y << 4;
    const int n0   = blockIdx.x << 6;

    const float* __restrict__ Ap = A + (size_t)(m0 + lm) * lda + (half << 1);
    const int nn0 = n0 + lm;
    const int nn1 = n0 + 16 + lm;
    const int nn2 = n0 + 32 + lm;
    const int nn3 = n0 + 48 + lm;
    const float* __restrict__ W0 = Wt + (size_t)nn0 * ldw + (half << 1);
    const float* __restrict__ W1 = Wt + (size_t)nn1 * ldw + (half << 1);
    const float* __restrict__ W2 = Wt + (size_t)nn2 * ldw + (half << 1);
    const float* __restrict__ W3 = Wt + (size_t)nn3 * ldw + (half << 1);

    v8f acc0 = {}, acc1 = {}, acc2 = {}, acc3 = {};

    for (int k0 = 0; k0 < K; k0 += 4) {
        const v2f a = *(const v2f*)(Ap + k0);
        v2f b0 = {}, b1 = {}, b2 = {}, b3 = {};
        if (nn0 < N) b0 = *(const v2f*)(W0 + k0);
        if (nn1 < N) b1 = *(const v2f*)(W1 + k0);
        if (nn2 < N) b2 = *(const v2f*)(W2 + k0);
        if (nn3 < N) b3 = *(const v2f*)(W3 + k0);
        acc0 = WMMA_F32(a, b0, acc0);
        acc1 = WMMA_F32(a, b1, acc1);
        acc2 = WMMA_F32(a, b2, acc2);
        acc3 = WMMA_F32(a, b3, acc3);
    }

    v8f accs[4] = {acc0, acc1, acc2, acc3};
    #pragma unroll
    for (int t = 0; t < 4; ++t) {
        const int n = n0 + (t << 4) + lm;
        if (n >= N) continue;
        float bv = 0.f;
        if (EPI != EPI_NONE) bv = bias[n];
        #pragma unroll
        for (int r = 0; r < 8; ++r) {
            const int m = m0 + r + (half << 3);
            Co[(size_t)m * ldc + n] = apply_epi(EPI, accs[t][r] + bv);
        }
    }
}

// ---------------------------------------------------------------------------
// LayerNorm over channels; x (B,C,H,W) -> xn token-major (B*H*W, C)
// ---------------------------------------------------------------------------
__global__ void ln_kernel(const float* __restrict__ x,
                          const float* __restrict__ g,
                          const float* __restrict__ bta,
                          float* __restrict__ xn)
{
    const int t = blockIdx.x * blockDim.x + threadIdx.x;
    if (t >= NTOK) return;
    const int b  = t >> 12;
    const int sp = t & (HWSZ - 1);
    const float* __restrict__ xp = x + (size_t)b * CH * HWSZ + sp;
    float mu = 0.f;
    for (int c = 0; c < CH; ++c) mu += xp[(size_t)c * HWSZ];
    mu *= (1.f / CH);
    float var = 0.f;
    for (int c = 0; c < CH; ++c) { float d = xp[(size_t)c * HWSZ] - mu; var += d * d; }
    var *= (1.f / CH);
    const float rstd = rsqrtf(var + 1e-5f);
    float* __restrict__ o = xn + (size_t)t * CH;
    for (int c = 0; c < CH; ++c)
        o[c] = (xp[(size_t)c * HWSZ] - mu) * rstd * g[c] + bta[c];
}

// ---------------------------------------------------------------------------
// Depthwise 3-tap conv along scan axis, with direction-specific reversal,
// emitting sequence-major u (NTOK, C). Reference conv is cross-correlation
// with zero pad (1,1): y[i] = k0*x[i-1] + k1*x[i] + k2*x[i+1] + bias.
// ---------------------------------------------------------------------------
__global__ void dwconv_seq_kernel(const float* __restrict__ xproj,
                                  const float* __restrict__ w3,
                                  const float* __restrict__ bs,
                                  float* __restrict__ u, int dir)
{
    const int idx = blockIdx.x * blockDim.x + threadIdx.x;
    if (idx >= NTOK * CH) return;
    const int c = idx & (CH - 1);
    const int t = idx >> 7;
    const int b = t >> 12;
    const int r = t & (HWSZ - 1);
    int i0 = -1, i1 = 0, i2 = -1;
    if (dir == 0) {                     // forward horizontal
        const int h = r >> 6, w = r & 63, base = (b << 12) + (h << 6);
        i1 = base + w;
        i0 = (w > 0)  ? base + w - 1 : -1;
        i2 = (w < 63) ? base + w + 1 : -1;
    } else if (dir == 1) {              // backward horizontal (w reversed)
        const int h = r >> 6, w = r & 63, wr = 63 - w, base = (b << 12) + (h << 6);
        i1 = base + wr;
        i0 = (wr < 63) ? base + wr + 1 : -1;
        i2 = (wr > 0)  ? base + wr - 1 : -1;
    } else if (dir == 2) {              // forward vertical (seq = b*W+w, l=h)
        const int w = r >> 6, h = r & 63;
        i1 = (b << 12) + (h << 6) + w;
        i0 = (h > 0)  ? (b << 12) + ((h - 1) << 6) + w : -1;
        i2 = (h < 63) ? (b << 12) + ((h + 1) << 6) + w : -1;
    } else {                            // backward vertical (h reversed)
        const int w = r >> 6, h = r & 63, hr = 63 - h;
        i1 = (b << 12) + (hr << 6) + w;
        i0 = (hr < 63) ? (b << 12) + ((hr + 1) << 6) + w : -1;
        i2 = (hr > 0)  ? (b << 12) + ((hr - 1) << 6) + w : -1;
    }
    const float k0 = w3[c * 3 + 0], k1 = w3[c * 3 + 1], k2 = w3[c * 3 + 2];
    float v = bs[c];
    if (i0 >= 0) v += k0 * xproj[(size_t)i0 * CH + c];
    v += k1 * xproj[(size_t)i1 * CH + c];
    if (i2 >= 0) v += k2 * xproj[(size_t)i2 * CH + c];
    u[(size_t)t * CH + c] = v;
}

// ---------------------------------------------------------------------------
// Causal depthwise conv1d (k=4, pad (3,0)) over xi (= xz[:, :DI]) + SiLU
// ---------------------------------------------------------------------------
__global__ void conv1d_silu_kernel(const float* __restrict__ xz,
                                   const float* __restrict__ cw,
                                   const float* __restrict__ cb,
                                   float* __restrict__ xs, int L)
{
    const int idx = blockIdx.x * blockDim.x + threadIdx.x;
    if (idx >= NTOK * DI) return;
    const int d = idx & (DI - 1);
    const int t = idx >> 8;
    const int pos = t & (L - 1);       // L is a power of two (4096 or 64)
    float acc = cb[d];
    const float* __restrict__ w = cw + d * 4;
    #pragma unroll
    for (int j = 0; j < 4; ++j) {
        const int l = pos - 3 + j;
        if (l >= 0) acc += w[j] * xz[(size_t)(t - 3 + j) * XZLD + d];
    }
    xs[(size_t)t * DI + d] = acc / (1.f + expf(-acc));   // silu
}

// ---------------------------------------------------------------------------
// Direct selective scan (used for L=64 vertical dirs).
// One thread per (sequence s, channel d); DS=16 state in registers.
// dt already softplus'ed. y = (sum_n h*C + x*D) * silu(z)
// ---------------------------------------------------------------------------
__global__ __launch_bounds__(256)
void scan_kernel(const float* __restrict__ dt,
                 const float* __restrict__ xdbl,
                 const float* __restrict__ xs,
                 const float* __restrict__ xz,
                 const float* __restrict__ Alog,
                 const float* __restrict__ Dp,
                 float* __restrict__ y, int L)
{
    const int s = blockIdx.x;
    const int d = threadIdx.x;
    float Ar[DS], h[DS];
    #pragma unroll
    for (int n = 0; n < DS; ++n) {
        Ar[n] = -expf(Alog[d * DS + n]);
        h[n]  = 0.f;
    }
    const float Dd = Dp[d];
    size_t t = (size_t)s * L;
    for (int l = 0; l < L; ++l, ++t) {
        const float dtv = dt[t * DI + d];
        const float xv  = xs[t * DI + d];
        const float dtx = dtv * xv;
        const float* __restrict__ Bp = xdbl + t * XD + DTR;      // wave-uniform
        const float* __restrict__ Cp = Bp + DS;                  // wave-uniform
        float yacc = 0.f;
        #pragma unroll
        for (int n = 0; n < DS; ++n) {
            const float hn = expf(dtv * Ar[n]) * h[n] + dtx * Bp[n];
            h[n] = hn;
            yacc += hn * Cp[n];
        }
        const float zr = xz[t * XZLD + DI + d];
        const float gate = zr / (1.f + expf(-zr));
        y[t * DI + d] = (yacc + xv * Dd) * gate;
    }
}

// ---------------------------------------------------------------------------
// Chunked parallel scan for L=4096 (3 phases). The linear recurrence
// h_t = dA_t * h_t-1 + b_t is associative per (d, n) state element:
//   phase 1: per 64-step chunk compute transfer P = prod(dA) and h_out (h_in=0)
//   phase 2: serial scan over chunk summaries; store each chunk's true h_in
//   phase 3: re-run each chunk seeded with h_in, producing gated y
// ---------------------------------------------------------------------------
__global__ __launch_bounds__(256)
void scan_chunk_phase1(const float* __restrict__ dt,
                       const float* __restrict__ xdbl,
                       const float* __restrict__ xs,
                       const float* __restrict__ Alog,
                       float* __restrict__ Pbuf,
                       float* __restrict__ Hbuf, int L, int nchunk)
{
    const int blk = blockIdx.x;             // s*nchunk + j
    const int s = blk / nchunk, j = blk - s * nchunk;
    const int d = threadIdx.x;
    float Ar[DS], h[DS], P[DS];
    #pragma unroll
    for (int n = 0; n < DS; ++n) {
        Ar[n] = -expf(Alog[d * DS + n]);
        h[n]  = 0.f;
        P[n]  = 1.f;
    }
    size_t t = (size_t)s * L + (size_t)j * CHUNK;
    for (int l = 0; l < CHUNK; ++l, ++t) {
        const float dtv = dt[t * DI + d];
        const float dtx = dtv * xs[t * DI + d];
        const float* __restrict__ Bp = xdbl + t * XD + DTR;
        #pragma unroll
        for (int n = 0; n < DS; ++n) {
            const float da = expf(dtv * Ar[n]);
            P[n] *= da;
            h[n]  = da * h[n] + dtx * Bp[n];
        }
    }
    float* __restrict__ pb = Pbuf + ((size_t)blk * DI + d) * DS;
    float* __restrict__ hb = Hbuf + ((size_t)blk * DI + d) * DS;
    #pragma unroll
    for (int n = 0; n < DS; ++n) { pb[n] = P[n]; hb[n] = h[n]; }
}

__global__ __launch_bounds__(256)
void scan_chunk_phase2(const float* __restrict__ Pbuf,
                       float* __restrict__ Hbuf, int nchunk)
{
    const int s = blockIdx.x;
    const int d = threadIdx.x;
    float hin[DS];
    #pragma unroll
    for (int n = 0; n < DS; ++n) hin[n] = 0.f;
    for (int j = 0; j < nchunk; ++j) {
        const size_t off = ((size_t)(s * nchunk + j) * DI + d) * DS;
        const float* __restrict__ pb = Pbuf + off;
        float* __restrict__ hb = Hbuf + off;
        #pragma unroll
        for (int n = 0; n < DS; ++n) {
            const float ho = hb[n];
            const float pp = pb[n];
            hb[n]  = hin[n];                 // chunk j's incoming state
            hin[n] = pp * hin[n] + ho;       // propagate to chunk j+1
        }
    }
}

__global__ __launch_bounds__(256)
void scan_chunk_phase3(const float* __restrict__ dt,
                       const float* __restrict__ xdbl,
                       const float* __restrict__ xs,
                       const float* __restrict__ xz,
                       const float* __restrict__ Alog,
                       const float* __restrict__ Dp,
                       const float* __restrict__ Hbuf,
                       float* __restrict__ y, int L, int nchunk)
{
    const int blk = blockIdx.x;
    const int s = blk / nchunk, j = blk - s * nchunk;
    const int d = threadIdx.x;
    float Ar[DS], h[DS];
    const float* __restrict__ hb = Hbuf + ((size_t)blk * DI + d) * DS;
    #pragma unroll
    for (int n = 0; n < DS; ++n) {
        Ar[n] = -expf(Alog[d * DS + n]);
        h[n]  = hb[n];
    }
    const float Dd = Dp[d];
    size_t t = (size_t)s * L + (size_t)j * CHUNK;
    for (int l = 0; l < CHUNK; ++l, ++t) {
        const float dtv = dt[t * DI + d];
        const float xv  = xs[t * DI + d];
        const float dtx = dtv * xv;
        const float* __restrict__ Bp = xdbl + t * XD + DTR;
        const float* __restrict__ Cp = Bp + DS;
        float yacc = 0.f;
        #pragma unroll
        for (int n = 0; n < DS; ++n) {
            const float hn = expf(dtv * Ar[n]) * h[n] + dtx * Bp[n];
            h[n] = hn;
            yacc += hn * Cp[n];
        }
        const float zr = xz[t * XZLD + DI + d];
        const float gate = zr / (1.f + expf(-zr));
        y[t * DI + d] = (yacc + xv * Dd) * gate;
    }
}

// ---------------------------------------------------------------------------
// Scatter sequence-major dir output back to spatial token-major accumulator
// ---------------------------------------------------------------------------
__global__ void scatter_kernel(const float* __restrict__ dirout,
                               float* __restrict__ hsum, int dir, int beta)
{
    const int idx = blockIdx.x * blockDim.x + threadIdx.x;
    if (idx >= NTOK * CH) return;
    const int c = idx & (CH - 1);
    const int t = idx >> 7;
    const int b = t >> 12;
    const int r = t & (HWSZ - 1);
    int sp;
    if (dir == 0)      sp = t;
    else if (dir == 1) { const int h = r >> 6, w = r & 63; sp = (b << 12) + (h << 6) + (63 - w); }
    else if (dir == 2) { const int w = r >> 6, h = r & 63; sp = (b << 12) + (h << 6) + w; }
    else               { const int w = r >> 6, h = r & 63; sp = (b << 12) + ((63 - h) << 6) + w; }
    const float v = dirout[(size_t)t * CH + c];
    float* p = hsum + (size_t)sp * CH + c;
    *p = beta ? (*p + v) : v;
}

__global__ void gate_kernel(const float* __restrict__ hsum,
                            const float* __restrict__ zg,
                            float* __restrict__ g)
{
    const int idx = blockIdx.x * blockDim.x + threadIdx.x;
    if (idx >= NTOK * CH) return;
    g[idx] = hsum[idx] * zg[idx];
}

// out[b,c,h,w] = clip(tmp[token, c] + x[b,c,h,w], -100, 100)
__global__ void final_kernel(const float* __restrict__ tmp,
                             const float* __restrict__ x,
                             float* __restrict__ out)
{
    const int e = blockIdx.x * blockDim.x + threadIdx.x;
    if (e >= BATCH * CH * HWSZ) return;
    const int w = e & 63;
    const int h = (e >> 6) & 63;
    const int c = (e >> 12) & (CH - 1);
    const int b = e >> 19;
    const int t = (b << 12) + (h << 6) + w;
    const float v = tmp[(size_t)t * CH + c] + x[e];
    out[e] = fminf(100.f, fmaxf(-100.f, v));
}

// ---------------------------------------------------------------------------
extern "C" void kernel_launch(void* const* d_in, const int* in_sizes, int n_in,
                              void* d_out, int out_size, void* d_ws, size_t ws_size,
                              hipStream_t stream)
{
    (void)in_sizes; (void)n_in; (void)out_size; (void)ws_size;

    const float* xin   = (const float*)d_in[0];
    const float* ln_g  = (const float*)d_in[1];
    const float* ln_b  = (const float*)d_in[2];
    const float* tox_w = (const float*)d_in[3];
    const float* tox_b = (const float*)d_in[4];
    const float* toz_w = (const float*)d_in[5];
    const float* toz_b = (const float*)d_in[6];
    const float* outw  = (const float*)d_in[15];
    const float* outb  = (const float*)d_in[16];

    // Workspace layout (floats). ~68.4 MB total.
    float* ws    = (float*)d_ws;
    float* xn    = ws;                    // (NTOK, C)   — later reused as dirout / gated
    float* xproj = xn    + (size_t)NTOK * CH;
    float* zgate = xproj + (size_t)NTOK * CH;
    float* hsum  = zgate + (size_t)NTOK * CH;
    float* u     = hsum  + (size_t)NTOK * CH;   // later reused as final-GEMM tmp
    float* xz    = u     + (size_t)NTOK * CH;   // (NTOK, 512)
    float* xs    = xz    + (size_t)NTOK * XZLD; // (NTOK, 256)
    float* xdbl  = xs    + (size_t)NTOK * DI;   // (NTOK, 40)
    float* dt    = xdbl  + (size_t)NTOK * XD;   // (NTOK, 256)
    float* ybuf  = dt    + (size_t)NTOK * DI;   // (NTOK, 256)
    float* Pbuf  = ybuf  + (size_t)NTOK * DI;   // (2*64, DI, DS) chunk transfers
    float* Hbuf  = Pbuf  + (size_t)2 * 64 * DI * DS; // (2*64, DI, DS) h_out/h_in
    float* dirout = xn;                          // reuse (xn dead after tox/toz)
    float* gated  = xn;                          // reuse (dirout dead after scatter)
    float* tmpo   = u;                           // reuse (u dead after last in_proj)

    const int EW = 256;                          // elementwise block size
    const dim3 gMN(2, NTOK / 32);                // N=128 GEMMs, 32-row tiles

    // 1) LayerNorm -> token-major xn
    ln_kernel<<<NTOK / EW, EW, 0, stream>>>(xin, ln_g, ln_b, xn);

    // 2) xproj = xn@tox^T + b ; zgate = gelu(xn@toz^T + b)
    gemm_wmma2_kernel<EPI_BIAS><<<gMN, 32, 0, stream>>>(
        xn, CH, tox_w, CH, tox_b, xproj, CH, CH);
    gemm_wmma2_kernel<EPI_GELU><<<gMN, 32, 0, stream>>>(
        xn, CH, toz_w, CH, toz_b, zgate, CH, CH);

    // 3) four directional Mamba branches
    for (int dir = 0; dir < 4; ++dir) {
        const float* cw     = (const float*)d_in[7 + 2 * dir];
        const float* cbias  = (const float*)d_in[8 + 2 * dir];
        const int base = 17 + 9 * dir;
        const float* in_w   = (const float*)d_in[base + 0];
        const float* c1d_w  = (const float*)d_in[base + 1];
        const float* c1d_b  = (const float*)d_in[base + 2];
        const float* xp_w   = (const float*)d_in[base + 3];
        const float* dt_w   = (const float*)d_in[base + 4];
        const float* dt_b   = (const float*)d_in[base + 5];
        const float* A_log  = (const float*)d_in[base + 6];
        const float* Dp     = (const float*)d_in[base + 7];
        const float* op_w   = (const float*)d_in[base + 8];

        const int L    = (dir < 2) ? HWSZ : HH;  // 4096 (horizontal) or 64 (vertical)
        const int nseq = NTOK / L;

        dwconv_seq_kernel<<<NTOK * CH / EW, EW, 0, stream>>>(xproj, cw, cbias, u, dir);

        // xz = u @ in_proj^T     (8192x128 @ 128x512)
        gemm_wmma2_kernel<EPI_NONE><<<dim3(8, NTOK / 32), 32, 0, stream>>>(
            u, CH, in_w, CH, nullptr, xz, XZLD, CH);

        conv1d_silu_kernel<<<NTOK * DI / EW, EW, 0, stream>>>(xz, c1d_w, c1d_b, xs, L);

        // x_dbl = xs @ x_proj^T  (8192x256 @ 256x40): only GEMM needing N guard
        gemm_wmma_guard_kernel<EPI_NONE><<<dim3(1, NTOK / 16), 32, 0, stream>>>(
            xs, DI, xp_w, DI, nullptr, xdbl, XD, XD, DI);

        // dt = softplus(x_dbl[:, :8] @ dt_proj^T + b)  (8192x8 @ 8x256)
        gemm_wmma2_kernel<EPI_SOFTPLUS><<<dim3(4, NTOK / 32), 32, 0, stream>>>(
            xdbl, XD, dt_w, DTR, dt_b, dt, DI, DTR);

        if (L == HWSZ) {
            // long scan: 3-phase chunked parallel scan (64 chunks of 64 steps)
            const int nchunk = HWSZ / CHUNK;     // 64
            scan_chunk_phase1<<<nseq * nchunk, DI, 0, stream>>>(
                dt, xdbl, xs, A_log, Pbuf, Hbuf, L, nchunk);
            scan_chunk_phase2<<<nseq, DI, 0, stream>>>(Pbuf, Hbuf, nchunk);
            scan_chunk_phase3<<<nseq * nchunk, DI, 0, stream>>>(
                dt, xdbl, xs, xz, A_log, Dp, Hbuf, ybuf, L, nchunk);
        } else {
            scan_kernel<<<nseq, DI, 0, stream>>>(dt, xdbl, xs, xz, A_log, Dp, ybuf, L);
        }

        // dirout = y @ out_proj^T (8192x256 @ 256x128)
        gemm_wmma2_kernel<EPI_NONE><<<gMN, 32, 0, stream>>>(
            ybuf, DI, op_w, DI, nullptr, dirout, CH, DI);

        scatter_kernel<<<NTOK * CH / EW, EW, 0, stream>>>(dirout, hsum, dir, dir == 0 ? 0 : 1);
    }

    // 4) gated = hsum * zgate ; tmp = gated @ out_w^T + out_b
    gate_kernel<<<NTOK * CH / EW, EW, 0, stream>>>(hsum, zgate, gated);
    gemm_wmma2_kernel<EPI_BIAS><<<gMN, 32, 0, stream>>>(
        gated, CH, outw, CH, outb, tmpo, CH, CH);

    // 5) out = clip(tmp + x, -100, 100), laid out (B,C,H,W)
    final_kernel<<<BATCH * CH * HWSZ / EW, EW, 0, stream>>>(tmpo, xin, (float*)d_out);
}